// SSIM3D_79199196938874
// MI455X (gfx1250) — compile-verified
//
#include <hip/hip_runtime.h>
#include <math.h>

typedef float v2f __attribute__((ext_vector_type(2)));
typedef float v8f __attribute__((ext_vector_type(8)));

#define NV 192          // volume edge
#define E  26           // ext tile edge (16 + 2*5 halo)
#define EP 28           // K padded to multiple of 4
// LDS layout (float offsets)
#define OFF_I1 0
#define SZ_EXT (E*E*EP)            /* 18928: raw ext cube [z26][y26][x28] */
#define OFF_I2 (OFF_I1 + SZ_EXT)   /* 18928                               */
#define OFF_B1 (OFF_I2 + SZ_EXT)   /* 37856: after X-conv [z28][y28][x16] */
#define SZ_B1  (EP*EP*16)          /* 12544 (rows 728..739 = dump area)   */
#define OFF_B2 (OFF_B1 + SZ_B1)    /* 50400: after Y-conv [z28][y16][x16] */
#define SZ_B2  (EP*16*16)          /* 7168                                */
#define OFF_RES (OFF_B2 + SZ_B2)   /* 57568: 5 blurred results            */
#define SZ_RES  (5*4096)           /* 20480                               */
#define OFF_WS  (OFF_RES + SZ_RES) /* 78048: 8 per-wave partials          */
#define LDS_FLOATS (OFF_WS + 8)    /* 78056 floats = 312224 B < 320 KB    */

__device__ __forceinline__ v8f wmma_f32_k4(v2f a, v2f b, v8f c) {
  // V_WMMA_F32_16X16X4_F32 : D = A(16x4) * B(4x16) + C(16x16), all f32
  return __builtin_amdgcn_wmma_f32_16x16x4_f32(
      false, a, false, b, (short)0, c, false, false);
}

// async global -> LDS copy (GLOBAL_LOAD_ASYNC_TO_LDS_B32, tracked by ASYNCcnt)
__device__ __forceinline__ void async_cp_b32(float* lds_dst, const float* gsrc) {
  asm volatile("global_load_async_to_lds_b32 %0, %1, off"
               :: "v"((unsigned)(size_t)lds_dst), "v"(gsrc)
               : "memory");
}
__device__ __forceinline__ void wait_asynccnt0() {
  asm volatile("s_wait_asynccnt 0x0" ::: "memory");
}

// ---- pass X specialized per quantity (q uniform -> full unswitch) ----------
// Q: 0=i1, 1=i2, 2=i1*i1, 3=i2*i2, 4=i1*i2
template <int Q>
__device__ __forceinline__ void pass_x(float* sm, int wid, int lane, int kh,
                                       int nn, int mw, const v2f* gg) {
  for (int t = wid; t < 43; t += 8) {             // ceil(676/16) M-tiles
    v8f c = {0, 0, 0, 0, 0, 0, 0, 0};
    int row = t * 16 + (lane & 15);
    int rowc = row < 676 ? row : 675;             // clamp pad rows (dumped)
    int zz = rowc / 26, yy = rowc % 26;
    const float* a1 = &sm[OFF_I1 + (zz * E + yy) * EP];
    const float* a2 = &sm[OFF_I2 + (zz * E + yy) * EP];
#pragma unroll
    for (int k = 0; k < 7; ++k) {
      int j = 4 * k + kh;
      v2f a;
      if constexpr (Q == 0) {
        a.x = a1[j]; a.y = a1[j + 1];
      } else if constexpr (Q == 1) {
        a.x = a2[j]; a.y = a2[j + 1];
      } else if constexpr (Q == 2) {
        float u = a1[j], v = a1[j + 1];
        a.x = u * u; a.y = v * v;
      } else if constexpr (Q == 3) {
        float u = a2[j], v = a2[j + 1];
        a.x = u * u; a.y = v * v;
      } else {
        a.x = a1[j] * a2[j]; a.y = a1[j + 1] * a2[j + 1];
      }
      c = wmma_f32_k4(a, gg[k], c);
    }
    // Write-back: the 8 rows rbase..rbase+7 are contiguous; carry (z,y)
    // incrementally -> one division per tile, cndmask+add per store.
    // Overflow rows (>=676) land in buf1 z==26 dump rows, never read.
    int rbase = t * 16 + mw;
    int zb = rbase / 26, yb = rbase % 26;
    int addr = OFF_B1 + (zb * EP + yb) * 16 + nn;
#pragma unroll
    for (int v = 0; v < 8; ++v) {
      sm[addr] = c[v];
      bool wrap = (yb == 25);
      addr += wrap ? 48 : 16;                     // +3 rows skips y-pad pair
      yb = wrap ? 0 : yb + 1;
    }
  }
}

__global__ void ssim3d_tile_kernel(const float* __restrict__ img1,
                                   const float* __restrict__ img2,
                                   float* __restrict__ partial) {
  extern __shared__ float sm[];
  const int tid  = threadIdx.x;
  const int lane = tid & 31;
  const int wid  = tid >> 5;

  const int tx = blockIdx.x, ty = blockIdx.y;
  const int batch = blockIdx.z / 12, tz = blockIdx.z % 12;
  const int x0 = tx * 16, y0 = ty * 16, z0 = tz * 16;
  const size_t base = (size_t)batch * NV * NV * NV;

  const int kh = (lane >> 4) << 1;   // K sub-offset for this half-wave (A & B)
  const int nn = lane & 15;          // N (and G-matrix row index) for this lane
  const int mw = (lane >> 4) << 3;   // +8 row offset for upper half-wave (C/D)

  // ---- per-lane Gaussian band operand, branchless, register-resident ------
  float S = 0.0f;
  for (int i = 0; i < 11; ++i) {
    float d = (float)(i - 5);
    S += __expf(-d * d * (1.0f / 4.5f));
  }
  const float invS = 1.0f / S;
  v2f gg[7];
#pragma unroll
  for (int k = 0; k < 7; ++k) {
    int j0 = 4 * k + kh - nn;
    int j1 = j0 + 1;
    float t0 = (float)(j0 - 5), t1 = (float)(j1 - 5);
    float e0 = __expf(-t0 * t0 * (1.0f / 4.5f)) * invS;
    float e1 = __expf(-t1 * t1 * (1.0f / 4.5f)) * invS;
    gg[k].x = ((unsigned)j0 <= 10u) ? e0 : 0.0f;
    gg[k].y = ((unsigned)j1 <= 10u) ? e1 : 0.0f;
  }

  // ---- zero buf1/buf2 once: pad rows must stay zero for the K=28 chains ---
  for (int i = tid; i < SZ_B1 + SZ_B2; i += 256) sm[OFF_B1 + i] = 0.0f;

  // ---- fill both raw ext cubes once via async global->LDS copies ----------
  for (int i = tid; i < E * EP; i += 256) {       // 728 (y,x) plane slots
    int y = i / EP, x = i - y * EP;
    int gy = y0 + y - 5, gx = x0 + x - 5;
    bool vyx = (x < E) && ((unsigned)gy < NV) && ((unsigned)gx < NV);
    size_t gplane = base + (size_t)gy * NV + gx;  // + gz*NV*NV per z
    for (int z = 0; z < E; ++z) {
      int gz = z0 + z - 5;
      bool ok = vyx && ((unsigned)gz < NV);
      int l1 = OFF_I1 + z * (E * EP) + i;
      int l2 = OFF_I2 + z * (E * EP) + i;
      if (ok) {
        size_t gi = gplane + (size_t)gz * (NV * NV);
        async_cp_b32(&sm[l1], &img1[gi]);
        async_cp_b32(&sm[l2], &img2[gi]);
      } else {
        sm[l1] = 0.0f;                            // 'same' conv zero padding
        sm[l2] = 0.0f;
      }
    }
  }
  wait_asynccnt0();
  __syncthreads();

  for (int q = 0; q < 5; ++q) {
    // ---- pass X: conv along x (unswitched on q) ---------------------------
    switch (q) {
      case 0: pass_x<0>(sm, wid, lane, kh, nn, mw, gg); break;
      case 1: pass_x<1>(sm, wid, lane, kh, nn, mw, gg); break;
      case 2: pass_x<2>(sm, wid, lane, kh, nn, mw, gg); break;
      case 3: pass_x<3>(sm, wid, lane, kh, nn, mw, gg); break;
      default: pass_x<4>(sm, wid, lane, kh, nn, mw, gg); break;
    }
    __syncthreads();

    // ---- pass Y: conv along y; A = G^T (registers), B = buf1 z-slice ------
    for (int z = wid; z < 26; z += 8) {
      v8f c = {0, 0, 0, 0, 0, 0, 0, 0};
#pragma unroll
      for (int k = 0; k < 7; ++k) {
        int j = 4 * k + kh;
        v2f b;
        b.x = sm[OFF_B1 + (z * EP + j) * 16 + nn];
        b.y = sm[OFF_B1 + (z * EP + j + 1) * 16 + nn];
        c = wmma_f32_k4(gg[k], b, c);
      }
#pragma unroll
      for (int v = 0; v < 8; ++v)
        sm[OFF_B2 + (z * 16 + v + mw) * 16 + nn] = c[v];
    }
    __syncthreads();

    // ---- pass Z: conv along z; A = G^T (registers), B = buf2 y-slice ------
    for (int y = wid; y < 16; y += 8) {
      v8f c = {0, 0, 0, 0, 0, 0, 0, 0};
#pragma unroll
      for (int k = 0; k < 7; ++k) {
        int j = 4 * k + kh;
        v2f b;
        b.x = sm[OFF_B2 + (j * 16 + y) * 16 + nn];
        b.y = sm[OFF_B2 + ((j + 1) * 16 + y) * 16 + nn];
        c = wmma_f32_k4(gg[k], b, c);
      }
#pragma unroll
      for (int v = 0; v < 8; ++v)
        sm[OFF_RES + q * 4096 + ((v + mw) * 16 + y) * 16 + nn] = c[v];
    }
    __syncthreads();
  }

  // ---- SSIM map + block reduction -----------------------------------------
  const float C1 = 1e-4f, C2 = 9e-4f;
  float local = 0.0f;
  for (int e = tid; e < 4096; e += 256) {
    float mu1 = sm[OFF_RES + e];
    float mu2 = sm[OFF_RES + 4096 + e];
    float b11 = sm[OFF_RES + 2 * 4096 + e];
    float b22 = sm[OFF_RES + 3 * 4096 + e];
    float b12 = sm[OFF_RES + 4 * 4096 + e];
    float m11 = mu1 * mu1, m22 = mu2 * mu2, m12 = mu1 * mu2;
    float s1 = b11 - m11, s2 = b22 - m22, s12 = b12 - m12;
    local += ((2.0f * m12 + C1) * (2.0f * s12 + C2)) /
             ((m11 + m22 + C1) * (s1 + s2 + C2));
  }
  for (int off = 16; off > 0; off >>= 1) local += __shfl_down(local, off);
  if (lane == 0) sm[OFF_WS + wid] = local;
  __syncthreads();
  if (tid == 0) {
    float s = 0.0f;
    for (int i = 0; i < 8; ++i) s += sm[OFF_WS + i];
    partial[blockIdx.x + 12 * blockIdx.y + 144 * blockIdx.z] = s;
  }
}

__global__ void reduce_mean_kernel(const float* __restrict__ partial,
                                   float* __restrict__ out,
                                   int n, float scale) {
  __shared__ float red[256];
  float s = 0.0f;
  for (int i = threadIdx.x; i < n; i += 256) s += partial[i];
  red[threadIdx.x] = s;
  __syncthreads();
  for (int k = 128; k > 0; k >>= 1) {
    if (threadIdx.x < k) red[threadIdx.x] += red[threadIdx.x + k];
    __syncthreads();
  }
  if (threadIdx.x == 0) out[0] = red[0] * scale;
}

extern "C" void kernel_launch(void* const* d_in, const int* in_sizes, int n_in,
                              void* d_out, int out_size, void* d_ws, size_t ws_size,
                              hipStream_t stream) {
  (void)in_sizes; (void)n_in; (void)out_size; (void)ws_size;
  const float* img1 = (const float*)d_in[0];
  const float* img2 = (const float*)d_in[1];
  float* out = (float*)d_out;
  float* partial = (float*)d_ws;            // 3456 floats

  dim3 grid(12, 12, 24);                    // 12^3 tiles x 2 batches
  ssim3d_tile_kernel<<<grid, 256, LDS_FLOATS * sizeof(float), stream>>>(
      img1, img2, partial);
  reduce_mean_kernel<<<1, 256, 0, stream>>>(
      partial, out, 3456, 1.0f / (2.0f * 192.0f * 192.0f * 192.0f));
}